// KinematicOperation_25082609008678
// MI455X (gfx1250) — compile-verified
//
#include <hip/hip_runtime.h>
#include <math.h>

// ---------------------------------------------------------------------------
// Kinematic forward pass for MI455X (gfx1250, wave32).
//   gen0: 2048 chains x 768 nodes   (jump at head, bonds after)
//   gen1: 2048 branches x 256 bonds seeded by gen0 transform at chain mid
// One wave per chain. Two-pass segmented prefix product:
//   pass1: per-lane serial product of a contiguous segment (24 / 8 mats)
//   scan : wave-level Hillis-Steele scan of 32 segment products. Combine =
//          batched independent 4x4 fp32 matmuls packed 4-at-a-time block-
//          diagonally into V_WMMA_F32_16X16X4_F32 (the only fp32 WMMA
//          shape), marshalled through LDS, updated IN PLACE (reads are
//          pre-staged to registers, so no ping-pong / copy-through needed).
//   pass2: re-walk segment from exclusive prefix, emit translation rows.
// Derived sincos values cached in LDS between passes (dofs rows are 36B /
// 4B-aligned, so global access is scalar; second pass runs entirely on LDS).
// ---------------------------------------------------------------------------

#define L0C 768
#define L1C 256
#define SEG0 (L0C / 32)   // 24
#define SEG1 (L1C / 32)   // 8
#define PI_F 3.14159265358979323846f

typedef float v2f __attribute__((ext_vector_type(2)));
typedef float v8f __attribute__((ext_vector_type(8)));

#if defined(__HIP_DEVICE_COMPILE__) && __has_builtin(__builtin_amdgcn_wmma_f32_16x16x4_f32)
#define USE_WMMA_F32X4 1
#else
#define USE_WMMA_F32X4 0
#endif

struct Mat4 { float m[4][4]; };

__device__ __forceinline__ void setIdent(Mat4& A) {
  #pragma unroll
  for (int i = 0; i < 4; ++i)
    #pragma unroll
    for (int j = 0; j < 4; ++j) A.m[i][j] = (i == j) ? 1.0f : 0.0f;
}

// C = A @ B for affine matrices (row 3 == [0,0,0,1]); 36 FMA instead of 64.
__device__ __forceinline__ Mat4 mulAffine(const Mat4& A, const Mat4& B) {
  Mat4 C;
  #pragma unroll
  for (int i = 0; i < 3; ++i) {
    const float a0 = A.m[i][0], a1 = A.m[i][1], a2 = A.m[i][2], a3 = A.m[i][3];
    #pragma unroll
    for (int j = 0; j < 4; ++j) {
      float acc = (j == 3) ? a3 : 0.0f;
      acc = fmaf(a0, B.m[0][j], acc);
      acc = fmaf(a1, B.m[1][j], acc);
      acc = fmaf(a2, B.m[2][j], acc);
      C.m[i][j] = acc;
    }
  }
  C.m[3][0] = 0.0f; C.m[3][1] = 0.0f; C.m[3][2] = 0.0f; C.m[3][3] = 1.0f;
  return C;
}

// bond = Rx(phi_p) @ Rz(pi - theta) @ Trans(d,0,0) @ Rx(phi_c), precomputed
// symbolically.  pre[] = {c1, s1, cz, sz, c3, s3, d, 0}
__device__ __forceinline__ void bondPre(float phi_p, float theta, float d,
                                        float phi_c, float pre[8]) {
  sincosf(phi_p, &pre[1], &pre[0]);
  const float a = PI_F - theta;
  sincosf(a, &pre[3], &pre[2]);
  sincosf(phi_c, &pre[5], &pre[4]);
  pre[6] = d;
  pre[7] = 0.0f;
}

__device__ __forceinline__ Mat4 bondFromPre(const float pre[8]) {
  const float c1 = pre[0], s1 = pre[1], cz = pre[2], sz = pre[3];
  const float c3 = pre[4], s3 = pre[5], d = pre[6];
  Mat4 M;
  M.m[0][0] = cz;       M.m[0][1] = -sz * c3;
  M.m[0][2] = sz * s3;  M.m[0][3] = cz * d;
  M.m[1][0] = c1 * sz;
  M.m[1][1] = fmaf(c1 * cz, c3, -s1 * s3);
  M.m[1][2] = fmaf(-c1 * cz, s3, -s1 * c3);
  M.m[1][3] = c1 * sz * d;
  M.m[2][0] = s1 * sz;
  M.m[2][1] = fmaf(s1 * cz, c3, c1 * s3);
  M.m[2][2] = fmaf(-s1 * cz, s3, c1 * c3);
  M.m[2][3] = s1 * sz * d;
  M.m[3][0] = 0.0f; M.m[3][1] = 0.0f; M.m[3][2] = 0.0f; M.m[3][3] = 1.0f;
  return M;
}

// R = Rz(c) @ Ry(b) @ Rx(a)
__device__ __forceinline__ void rot3(float a, float b, float c, float R[3][3]) {
  float sa, ca, sb, cb, sc, cc;
  sincosf(a, &sa, &ca);
  sincosf(b, &sb, &cb);
  sincosf(c, &sc, &cc);
  R[0][0] = cc * cb; R[0][1] = fmaf(cc * sb, sa, -sc * ca); R[0][2] = fmaf(cc * sb, ca, sc * sa);
  R[1][0] = sc * cb; R[1][1] = fmaf(sc * sb, sa, cc * ca);  R[1][2] = fmaf(sc * sb, ca, -cc * sa);
  R[2][0] = -sb;     R[2][1] = cb * sa;                     R[2][2] = cb * ca;
}

// jump = Trans(d0,d1,d2) @ rot(d3,d4,d5) @ rot(d6,d7,d8)
__device__ __forceinline__ Mat4 jumpMat(const float d9[9]) {
  float R1[3][3], R2[3][3];
  rot3(d9[3], d9[4], d9[5], R1);
  rot3(d9[6], d9[7], d9[8], R2);
  Mat4 M;
  #pragma unroll
  for (int i = 0; i < 3; ++i)
    #pragma unroll
    for (int j = 0; j < 3; ++j)
      M.m[i][j] = fmaf(R1[i][0], R2[0][j],
                  fmaf(R1[i][1], R2[1][j], R1[i][2] * R2[2][j]));
  M.m[0][3] = d9[0]; M.m[1][3] = d9[1]; M.m[2][3] = d9[2];
  M.m[3][0] = 0.0f; M.m[3][1] = 0.0f; M.m[3][2] = 0.0f; M.m[3][3] = 1.0f;
  return M;
}

// ---------------------------------------------------------------------------
// One in-place Hillis-Steele scan step over 32 matrices held in LDS
// (row-major 4x4, 16 floats each):
//   buf[l] <- (l >= DELTA) ? buf[l-DELTA] @ buf[l] : buf[l]   (unchanged)
// All A/B fragments are pre-staged to registers, then a barrier, then the
// WMMAs write back in place -- no ping-pong buffer, no copy-through.
// WMMA: V_WMMA_F32_16X16X4_F32 computes D(16x16) = A(16x4) @ B(4x16);
// stacking 4 left matrices in A and 4 right matrices in B yields the 4
// wanted products on the diagonal blocks of D (32 matrices = <=8 WMMAs).
// Groups whose 4 targets are all < DELTA are dropped at compile time.
// A layout: lane m<16 holds A[m][0..1] in vgpr0/1, lane m+16 holds A[m][2..3].
// B layout: lane n<16 holds B[0..1][n],     lane n+16 holds B[2..3][n].
// D layout: vgpr v: lanes 0-15 = D[v][n], lanes 16-31 = D[v+8][n].
// ---------------------------------------------------------------------------
#if USE_WMMA_F32X4
template <int DELTA>
__device__ __forceinline__ void scan_step(float* buf, int lane) {
  const int  m      = lane & 15;
  const int  blk    = m >> 2;          // which 4x4 block this lane serves
  const int  r      = m & 3;           // row-in-block (A) / col-in-block (B)
  const bool hiHalf = (lane >= 16);
  const int  k0     = hiHalf ? 2 : 0;  // K indices owned by this half-wave
  // Diagonal block blk of D: rows 4*blk..4*blk+3 live in the low half for
  // blk<2 and in the high half for blk>=2, at vgprs (blk&1)*4 .. +3.
  const bool writer = hiHalf ? (blk >= 2) : (blk < 2);
  const bool useHi  = (blk & 1) != 0;
  v2f a[8], b[8];
  #pragma unroll
  for (int g = 0; g < 8; ++g) {
    if (4 * g + 3 < DELTA) continue;   // whole group unchanged this step
    const int l = (g << 2) + blk;      // right-matrix / target index
    int la = l - DELTA;                // left-matrix index
    if (la < 0) la = 0;                // keep EXEC full; result discarded
    a[g].x = buf[la * 16 + r * 4 + k0];
    a[g].y = buf[la * 16 + r * 4 + k0 + 1];
    b[g].x = buf[l * 16 + k0 * 4 + r];
    b[g].y = buf[l * 16 + (k0 + 1) * 4 + r];
  }
  __syncthreads();                     // all reads staged before any write
  #pragma unroll
  for (int g = 0; g < 8; ++g) {
    if (4 * g + 3 < DELTA) continue;
    const int l = (g << 2) + blk;
    v8f c = {0.f, 0.f, 0.f, 0.f, 0.f, 0.f, 0.f, 0.f};
    v8f d = __builtin_amdgcn_wmma_f32_16x16x4_f32(
        false, a[g], false, b[g], (short)0, c, false, false);
    if (writer && l >= DELTA) {
      #pragma unroll
      for (int i = 0; i < 4; ++i) {
        // constant-index extracts + one select (avoid dynamic v8f indexing)
        const float lo = d[i], hi = d[i + 4];
        buf[l * 16 + i * 4 + r] = useHi ? hi : lo;
      }
    }
  }
}
#else
template <int DELTA>
__device__ __forceinline__ void scan_step(float* buf, int lane) {
  Mat4 Lm, Rm;
  #pragma unroll
  for (int i = 0; i < 4; ++i)
    #pragma unroll
    for (int j = 0; j < 4; ++j) {
      Rm.m[i][j] = buf[lane * 16 + i * 4 + j];
      const int la = (lane >= DELTA) ? (lane - DELTA) : 0;
      Lm.m[i][j] = buf[la * 16 + i * 4 + j];
    }
  __syncthreads();
  if (lane >= DELTA) {
    const Mat4 Q = mulAffine(Lm, Rm);
    #pragma unroll
    for (int i = 0; i < 4; ++i)
      #pragma unroll
      for (int j = 0; j < 4; ++j) buf[lane * 16 + i * 4 + j] = Q.m[i][j];
  }
}
#endif

// Exclusive left-to-right scan of the 32 per-lane segment products.
// smem must hold 512 floats.
__device__ __forceinline__ Mat4 wave_scan_excl(float* smem, const Mat4& P,
                                               int lane) {
  #pragma unroll
  for (int i = 0; i < 4; ++i)
    #pragma unroll
    for (int j = 0; j < 4; ++j) smem[lane * 16 + i * 4 + j] = P.m[i][j];
  __syncthreads();
  scan_step<1>(smem, lane);  __syncthreads();
  scan_step<2>(smem, lane);  __syncthreads();
  scan_step<4>(smem, lane);  __syncthreads();
  scan_step<8>(smem, lane);  __syncthreads();
  scan_step<16>(smem, lane); __syncthreads();     // inclusive result in smem
  Mat4 E;
  if (lane == 0) {
    setIdent(E);
  } else {
    const float* q = smem + (lane - 1) * 16;
    #pragma unroll
    for (int i = 0; i < 4; ++i)
      #pragma unroll
      for (int j = 0; j < 4; ++j) E.m[i][j] = q[i * 4 + j];
  }
  return E;
}

// ---------------------------------------------------------------------------
// gen0: one wave per chain of 768 nodes (jump head + 767 bonds).
// ---------------------------------------------------------------------------
__global__ void __launch_bounds__(32)
k_gen0(const float* __restrict__ dofs, const int* __restrict__ g0,
       const int* __restrict__ idi, float* __restrict__ out,
       float* __restrict__ wsRoots) {
  __shared__ __align__(16) float sdof[L0C * 8];   // cached bond sincos (24KB)
  __shared__ __align__(16) float sm[512];         // scan buffer (2KB)
  const int c    = blockIdx.x;
  const int lane = threadIdx.x;
  const int* path = g0 + c * (L0C + 1) + 1;       // skip node 0 (identity)
  const int  jb   = lane * SEG0;

  // ---- pass 1: per-lane serial segment product -------------------------
  Mat4 P; setIdent(P);
  #pragma unroll 1
  for (int t = 0; t < SEG0; ++t) {
    const int    j    = jb + t;
    const int    node = path[j];
    const float* dr   = dofs + (size_t)node * 9;
    __builtin_prefetch(dr + 18, 0, 0);            // global_prefetch_b8 ahead
    Mat4 M;
    if (j == 0) {                                 // jump head (doftype==1)
      float d9[9];
      #pragma unroll
      for (int k = 0; k < 9; ++k) d9[k] = dr[k];
      M = jumpMat(d9);
    } else {
      float pre[8];
      bondPre(dr[0], dr[1], dr[2], dr[3], pre);
      *(float4*)&sdof[j * 8]     = make_float4(pre[0], pre[1], pre[2], pre[3]);
      *(float4*)&sdof[j * 8 + 4] = make_float4(pre[4], pre[5], pre[6], pre[7]);
      M = bondFromPre(pre);
    }
    P = mulAffine(P, M);
  }

  // ---- wave scan (WMMA) -> exclusive prefix ----------------------------
  Mat4 run = wave_scan_excl(sm, P, lane);

  // ---- pass 2: apply prefix, emit coords, stash branch-root transform --
  #pragma unroll 1
  for (int t = 0; t < SEG0; ++t) {
    const int j    = jb + t;
    const int node = path[j];
    Mat4 M;
    if (j == 0) {                                  // the jump head
      const float* dr = dofs + (size_t)node * 9;
      float d9[9];
      #pragma unroll
      for (int k = 0; k < 9; ++k) d9[k] = dr[k];
      M = jumpMat(d9);
    } else {
      const float4 u0 = *(const float4*)&sdof[j * 8];
      const float4 u1 = *(const float4*)&sdof[j * 8 + 4];
      const float pre[8] = {u0.x, u0.y, u0.z, u0.w, u1.x, u1.y, u1.z, u1.w};
      M = bondFromPre(pre);
    }
    run = mulAffine(run, M);
    const int o = idi[node - 1];
    float* op = out + (size_t)o * 3;
    op[0] = run.m[0][3]; op[1] = run.m[1][3]; op[2] = run.m[2][3];
    if (j == (L0C / 2)) {                          // branch root for gen1
      float* w = wsRoots + (size_t)c * 16;
      #pragma unroll
      for (int i = 0; i < 4; ++i)
        #pragma unroll
        for (int k = 0; k < 4; ++k) w[i * 4 + k] = run.m[i][k];
    }
  }
}

// ---------------------------------------------------------------------------
// gen1: one wave per branch of 256 bond nodes, seeded by the stored root.
// ---------------------------------------------------------------------------
__global__ void __launch_bounds__(32)
k_gen1(const float* __restrict__ dofs, const int* __restrict__ g1,
       const int* __restrict__ idi, float* __restrict__ out,
       const float* __restrict__ wsRoots) {
  __shared__ __align__(16) float sdof[L1C * 8];   // 8KB
  __shared__ __align__(16) float sm[512];         // 2KB
  const int c    = blockIdx.x;
  const int lane = threadIdx.x;
  const int* path = g1 + c * (L1C + 1) + 1;       // skip the root entry
  const int  jb   = lane * SEG1;

  Mat4 P; setIdent(P);
  #pragma unroll 1
  for (int t = 0; t < SEG1; ++t) {
    const int    j    = jb + t;
    const int    node = path[j];
    const float* dr   = dofs + (size_t)node * 9;
    __builtin_prefetch(dr + 18, 0, 0);
    float pre[8];
    bondPre(dr[0], dr[1], dr[2], dr[3], pre);
    *(float4*)&sdof[j * 8]     = make_float4(pre[0], pre[1], pre[2], pre[3]);
    *(float4*)&sdof[j * 8 + 4] = make_float4(pre[4], pre[5], pre[6], pre[7]);
    P = mulAffine(P, bondFromPre(pre));
  }

  Mat4 E = wave_scan_excl(sm, P, lane);

  // Premultiply by the branch root's global transform (wave-uniform load).
  Mat4 R;
  {
    const float* w = wsRoots + (size_t)c * 16;
    #pragma unroll
    for (int i = 0; i < 4; ++i)
      #pragma unroll
      for (int k = 0; k < 4; ++k) R.m[i][k] = w[i * 4 + k];
  }
  Mat4 run = mulAffine(R, E);

  #pragma unroll 1
  for (int t = 0; t < SEG1; ++t) {
    const int j    = jb + t;
    const int node = path[j];
    const float4 u0 = *(const float4*)&sdof[j * 8];
    const float4 u1 = *(const float4*)&sdof[j * 8 + 4];
    const float pre[8] = {u0.x, u0.y, u0.z, u0.w, u1.x, u1.y, u1.z, u1.w};
    run = mulAffine(run, bondFromPre(pre));
    const int o = idi[node - 1];
    float* op = out + (size_t)o * 3;
    op[0] = run.m[0][3]; op[1] = run.m[1][3]; op[2] = run.m[2][3];
  }
}

extern "C" void kernel_launch(void* const* d_in, const int* in_sizes, int n_in,
                              void* d_out, int out_size, void* d_ws,
                              size_t ws_size, hipStream_t stream) {
  const float* dofs    = (const float*)d_in[0];
  const int*   g0      = (const int*)d_in[2];
  const int*   g1      = (const int*)d_in[3];
  const int*   idi     = (const int*)d_in[4];
  float*       out     = (float*)d_out;
  float*       wsRoots = (float*)d_ws;             // 2048 * 16 floats = 128KB

  const int nC0 = in_sizes[2] / (L0C + 1);         // 2048
  const int nC1 = in_sizes[3] / (L1C + 1);         // 2048

  k_gen0<<<dim3(nC0), dim3(32), 0, stream>>>(dofs, g0, idi, out, wsRoots);
  k_gen1<<<dim3(nC1), dim3(32), 0, stream>>>(dofs, g1, idi, out, wsRoots);
  (void)n_in; (void)out_size; (void)ws_size;
}